// SimpleCarCost_1039382085788
// MI455X (gfx1250) — compile-verified
//
#include <hip/hip_runtime.h>
#include <hip/hip_bf16.h>
#include <stdint.h>

// ---- constants mirroring the reference cost_cfg ----
#define CAR_W2            1.0f           // 2.0/2
#define CAR_L2            2.0f           // 4.0/2
#define SOFT_SPEED_LIM    10.0f
#define FALL_OFF_PENALTY  1000.0f
#define FLOOR_ELEVATION   0.1f
#define MAP_RES           0.05f
#define INV_MAP_RES       20.0f
#define MAP_SIZE          2048
#define MAP_ORIGIN        (-51.2f)       // -MAP_SIZE*MAP_RES/2

#define BLOCK 256

// Bilinear sample of one env's heightmap at world (x,y).
// Base pointer is wave-uniform -> saddr + 32-bit voffset addressing.
__device__ __forceinline__ float bilin(const float* __restrict__ hme, float x, float y) {
    float fx = (x - MAP_ORIGIN) * INV_MAP_RES;
    float fy = (y - MAP_ORIGIN) * INV_MAP_RES;
    int x0 = (int)floorf(fx);
    int y0 = (int)floorf(fy);
    x0 = min(max(x0, 0), MAP_SIZE - 2);
    y0 = min(max(y0, 0), MAP_SIZE - 2);
    float tx = fminf(fmaxf(fx - (float)x0, 0.0f), 1.0f);
    float ty = fminf(fmaxf(fy - (float)y0, 0.0f), 1.0f);
    unsigned idx = (unsigned)y0 * (unsigned)MAP_SIZE + (unsigned)x0;  // < 2^22
    float h00 = hme[idx];
    float h01 = hme[idx + 1];
    float h10 = hme[idx + MAP_SIZE];
    float h11 = hme[idx + MAP_SIZE + 1];
    // lerp form == reference's 4-weight blend
    float a = __builtin_fmaf(tx, h01 - h00, h00);
    float b = __builtin_fmaf(tx, h11 - h10, h10);
    return __builtin_fmaf(ty, b - a, a);
}

__device__ __forceinline__ float corner_h(const float* __restrict__ hme, float x, float y) {
    float h = bilin(hme, x, y);
    return (h != h) ? FALL_OFF_PENALTY : h;   // nan_to_num(nan=1000)
}

__global__ __launch_bounds__(BLOCK)
void car_cost_kernel(const float* __restrict__ state,
                     const float* __restrict__ hm,
                     float* __restrict__ out,
                     int N, int KT) {
    // 256 records x 12 floats, staged linearly; b128 async ops need 16B alignment.
    __shared__ __align__(16) float smem[BLOCK * 12];

    const int tid = threadIdx.x;
    const int e   = blockIdx.y;                                 // env index: uniform, free
    const long long recBase = (long long)e * KT + (long long)blockIdx.x * BLOCK;

    // Wave-uniform 64-bit base (SGPR pair) + per-lane 32-bit byte offset (GVS form).
    const float* gblock = state + recBase * 12;
    // Last legal 16B-aligned offset within the state array, relative to gblock.
    const unsigned maxOff = (unsigned)(((long long)N - recBase) * 48 - 16);

    // ---- async global -> LDS staging (CDNA5 ASYNCcnt path), fully coalesced ----
#pragma unroll
    for (int j = 0; j < 3; ++j) {
        unsigned off = (unsigned)(tid + j * BLOCK) * 16u;       // byte offset
        off = min(off, maxOff);                                 // tail clamp (dup loads OK)
        unsigned ldsOff = (unsigned)(uintptr_t)(&smem[(tid + j * BLOCK) * 4]);
        asm volatile("global_load_async_to_lds_b128 %0, %1, %2"
                     :: "v"(ldsOff), "v"(off), "s"(gblock)
                     : "memory");
    }
    asm volatile("s_wait_asynccnt 0x0" ::: "memory");  // my wave's async copies done
    __syncthreads();                                   // all 8 waves' copies visible

    const int local = blockIdx.x * BLOCK + tid;
    if (local >= KT) return;

    const float* rec = &smem[tid * 12];
    const float x   = rec[0];
    const float y   = rec[1];
    const float yaw = rec[5];
    const float vx  = rec[6];
    const float vy  = rec[7];

    const float* hme = hm + (size_t)e * ((size_t)MAP_SIZE * MAP_SIZE);

    // velocity cost
    float v = sqrtf(__builtin_fmaf(vx, vx, vy * vy));
    float vel_cost = fabsf(SOFT_SPEED_LIM - v);

    // center sample + floor penalty
    float gz = bilin(hme, x, y);
    float fall_off = (gz <= FLOOR_ELEVATION) ? FALL_OFF_PENALTY : gz;

    // footprint corners: hardware transcendentals (v_sin_f32/v_cos_f32 co-execute
    // with VALU on CDNA5; yaw is O(1) rad so fast range scaling is plenty accurate)
    float s, c;
    __sincosf(yaw, &s, &c);
    const float lx = CAR_L2 * c, ly = CAR_L2 * s;   // forward vector * L/2
    const float wx = CAR_W2 * s, wy = CAR_W2 * c;   // lateral components
    float fp = corner_h(hme, x + lx - wx, y + ly + wy)   // fl
             + corner_h(hme, x + lx + wx, y + ly - wy)   // fr
             + corner_h(hme, x - lx - wx, y - ly + wy)   // bl
             + corner_h(hme, x - lx + wx, y - ly - wy);  // br

    out[recBase + tid] = fall_off + fp + vel_cost;
}

extern "C" void kernel_launch(void* const* d_in, const int* in_sizes, int n_in,
                              void* d_out, int out_size, void* d_ws, size_t ws_size,
                              hipStream_t stream) {
    const float* state = (const float*)d_in[0];
    // d_in[1] (controls) is unused by the reference output
    const float* hm = (const float*)d_in[2];
    float* out = (float*)d_out;

    const long long N = (long long)in_sizes[0] / 12;                 // E*K*T
    int E = in_sizes[2] / (MAP_SIZE * MAP_SIZE);
    if (E < 1) E = 1;
    const int KT = (int)(N / E);

    dim3 grid((unsigned)((KT + BLOCK - 1) / BLOCK), (unsigned)E, 1);
    car_cost_kernel<<<grid, BLOCK, 0, stream>>>(state, hm, out, (int)N, KT);
}